// CPdecomposition_2164663517942
// MI455X (gfx1250) — compile-verified
//
#include <hip/hip_runtime.h>
#include <math.h>

// CP-decomposition (TensoRF-style) forward for MI455X / gfx1250.
//
// - x-interp collapses to a 4608-entry LDS pair table (36 KB) since ix/wx
//   depend only on the channel index.
// - Per-ray work: 6 ds_load_2addr_b64 + lerp-product per component pair.
// - 128->8 component fold done with V_WMMA_F32_16X16X4_F32 (fp32, exact)
//   against constant 0/1 selection matrices (alternating even/odd cols).
// - Inner loop is gather-then-compute so all LDS loads for a WMMA step
//   issue back-to-back before one s_wait_dscnt, hiding DS latency.

typedef __attribute__((ext_vector_type(2))) float v2f;
typedef __attribute__((ext_vector_type(8))) float v8f;

#define N_RAYS 131072
#define IN_CH  6
#define WIDTH  512
#define NCOMPT 128   // N_COMP * OUT_CH
#define OUT_CH 8

__global__ __launch_bounds__(256) void cp_forward_kernel(
    const float* __restrict__ x,      // (N_RAYS, 6)
    const float* __restrict__ plane,  // (128, 6, 512)
    float* __restrict__ out)          // (N_RAYS, 8)
{
    // Pair table: sP[k*36 + c*6 + i] = ( T(k,c,i), T(k,c,min(i+1,5)) )
    // where T(k,c,i) = plane[k,i,j0_c]*(1-wx_c) + plane[k,i,j1_c]*wx_c
    __shared__ float2 sP[NCOMPT * IN_CH * 6];   // 36 KB

    const int tid = threadIdx.x;

    // ---- Phase 1: build the x-collapsed pair table (block-cooperative) ----
    for (int idx = tid; idx < NCOMPT * IN_CH * 6; idx += 256) {
        int k = idx / 36;
        int r = idx - k * 36;
        int c = r / 6;
        int i = r - c * 6;
        float h   = -1.0f + 0.4f * (float)c;              // linspace(-1,1,6)
        float ix  = (h + 1.0f) * 0.5f * 511.0f;
        float j0f = floorf(ix);
        float wx  = ix - j0f;
        int j0 = (int)j0f; j0 = j0 < 0 ? 0 : (j0 > WIDTH - 1 ? WIDTH - 1 : j0);
        int j1 = j0 + 1 > WIDTH - 1 ? WIDTH - 1 : j0 + 1;
        int i1 = i + 1 > 5 ? 5 : i + 1;
        const float* r0 = plane + (k * IN_CH + i)  * WIDTH;
        const float* r1 = plane + (k * IN_CH + i1) * WIDTH;
        float t0 = r0[j0] * (1.0f - wx) + r0[j1] * wx;
        float t1 = r1[j0] * (1.0f - wx) + r1[j1] * wx;
        sP[idx] = make_float2(t0, t1);
    }
    __syncthreads();

    // ---- Phase 2: per-ray sampling + WMMA component fold ----
    const int  lane = tid & 31;
    const int  wave = tid >> 5;
    const int  lrow = lane & 15;        // A-matrix row = ray-in-wave; also N col for C/D
    const bool lolane = lane < 16;
    const int  rayBase = (blockIdx.x * 8 + wave) * 16;
    const int  ray = rayBase + lrow;

    // y coordinate per channel (depends on the ray); precompute LDS offsets.
    const float* xr = x + ray * IN_CH;
    float wy[IN_CH];
    int   off[IN_CH];                   // c*6 + i0[c], invariant over t
#pragma unroll
    for (int c = 0; c < IN_CH; ++c) {
        float norm = xr[c] * 2.0f - 1.0f;
        float iy   = (norm + 1.0f) * 0.5f * 5.0f;   // (H-1) = 5
        float f0   = floorf(iy);
        wy[c] = iy - f0;
        int ii = (int)f0;
        ii = ii < 0 ? 0 : (ii > 5 ? 5 : ii);
        off[c] = c * 6 + ii;
    }

    // Constant 0/1 selection matrices for B (4x16, f32):
    //   B layout: VGPR0 = row K (0 for lanes 0-15, 2 for lanes 16-31),
    //             VGPR1 = row K+1; column N = lane%16.
    //   even t:  B[kk, n] = (n == kk)        -> cols 0..3
    //   odd  t:  B[kk, n] = (n == kk + 4)    -> cols 4..7
    const int kk0 = lolane ? 0 : 2;
    v2f Beven, Bodd;
    Beven.x = (lrow == kk0)     ? 1.0f : 0.0f;
    Beven.y = (lrow == kk0 + 1) ? 1.0f : 0.0f;
    Bodd.x  = (lrow == kk0 + 4) ? 1.0f : 0.0f;
    Bodd.y  = (lrow == kk0 + 5) ? 1.0f : 0.0f;

    v8f acc = {};   // C/D accumulator: out[ray, o] partial sums

    // A-matrix layout for 16x4 f32: lanes 0-15 hold K=4t+0 (A.x), 4t+1 (A.y);
    // lanes 16-31 hold K=4t+2, 4t+3 for the same 16 rays.
    const int koff = lolane ? 0 : 2;
#pragma unroll 4
    for (int t = 0; t < 32; ++t) {
        const float2* p0 = &sP[(4 * t + koff) * (IN_CH * 6)];

        // Gather phase: issue all 12 LDS pair loads for this step up front
        // (backend emits 6x ds_load_2addr_b64, then a single dscnt wait).
        float2 q0[IN_CH], q1[IN_CH];
#pragma unroll
        for (int c = 0; c < IN_CH; ++c) {
            q0[c] = p0[off[c]];                     // component 4t+koff
            q1[c] = p0[(IN_CH * 6) + off[c]];       // component 4t+koff+1
        }

        // Compute phase: lerp + product chain (pairs -> v_pk_fma / v_dual_*)
        float f0 = 1.0f, f1 = 1.0f;
#pragma unroll
        for (int c = 0; c < IN_CH; ++c) {
            f0 *= q0[c].x + wy[c] * (q0[c].y - q0[c].x);
            f1 *= q1[c].x + wy[c] * (q1[c].y - q1[c].x);
        }

        v2f A; A.x = f0; A.y = f1;
        // D = A(16x4) * B(4x16) + C : accumulates feat into out columns 0..7
        acc = __builtin_amdgcn_wmma_f32_16x16x4_f32(
                  false, A, false, (t & 1) ? Bodd : Beven,
                  (short)0, acc, false, false);
    }

    // ---- Phase 3: sigmoid + store ----
    // C/D layout: VGPR j -> row j (lanes 0-15) or row j+8 (lanes 16-31), col = lane%16.
    if (lrow < OUT_CH) {
#pragma unroll
        for (int j = 0; j < 8; ++j) {
            int row = lolane ? j : (j + 8);
            float z = acc[j];
            out[(rayBase + row) * OUT_CH + lrow] = 1.0f / (1.0f + __expf(-z));
        }
    }
}

extern "C" void kernel_launch(void* const* d_in, const int* in_sizes, int n_in,
                              void* d_out, int out_size, void* d_ws, size_t ws_size,
                              hipStream_t stream) {
    const float* x     = (const float*)d_in[0];   // (131072, 6)
    const float* plane = (const float*)d_in[1];   // (128, 6, 512)
    float* out = (float*)d_out;                   // (131072, 8)

    // 256 threads = 8 waves; each wave covers 16 rays -> 128 rays/block
    const int blocks = N_RAYS / 128;              // 1024, exact (no tail)
    cp_forward_kernel<<<blocks, 256, 0, stream>>>(x, plane, out);
}